// PathAbsorbing_80418967650384
// MI455X (gfx1250) — compile-verified
//
#include <hip/hip_runtime.h>
#include <stdint.h>

#define G 12
#define WCELLS 144          // G*G
#define PROBS_PER_BLOCK 16  // 8 waves * 2 problems/wave
#define THREADS 256
#define PINF 1.0e9f

// LDS layout (bytes), all 16B aligned
#define STAGE_OFF 0                     // 16 * 144 * 4 = 9216  (w indices)
#define DIST_OFF  9216                  // 16 * 196 * 4 = 12544 (14x14 padded dist)
#define MASK_OFF  (9216 + 12544)        // 16 * 144 * 4 = 9216  (path mask)
#define LDS_BYTES (MASK_OFF + 9216)

typedef __attribute__((address_space(3))) char lds_char_t;

// byte offset of an LDS object relative to the workgroup LDS base
// (generic->local addrspacecast keeps only the low 32-bit DS address)
__device__ __forceinline__ unsigned lds_lo(void* p) {
    return (unsigned)(unsigned long long)(lds_char_t*)p;
}

__global__ __launch_bounds__(THREADS)
void PathAbsorbing_kernel(const int* __restrict__ w,
                          const float* __restrict__ costs,
                          int* __restrict__ out, int K) {
    extern __shared__ char smem[];
    int*   stage = (int*)(smem + STAGE_OFF);
    float* dpad  = (float*)(smem + DIST_OFF);
    int*   mlds  = (int*)(smem + MASK_OFF);

    const int tid    = threadIdx.x;
    const int lane   = tid & 31;
    const int waveId = tid >> 5;
    const int r      = lane & 15;        // row handled by this lane (12..15 idle)
    const int half   = lane >> 4;        // which problem within the wave
    const int pLoc   = waveId * 2 + half;            // 0..15 within block
    const int k      = blockIdx.x * PROBS_PER_BLOCK + pLoc;
    const bool active = (r < G) && (k < K);

    // ---- async stage: this block's w indices (2304 dwords = 576 x b128) into LDS ----
    // Branch-free tail handling: clamp the global byte offset (clamped chunks land in
    // staging slots of problems that are never consumed).
    {
        const unsigned long long gbase = (unsigned long long)w;
        const unsigned totalBytes = (unsigned)K * (WCELLS * 4);
        const unsigned maxOff     = totalBytes - 16;
        const unsigned blockByte  = (unsigned)blockIdx.x * (PROBS_PER_BLOCK * WCELLS * 4);
        const unsigned ldsBase    = lds_lo(stage);

        unsigned o0 = blockByte + (unsigned)tid * 16;
        o0 = (o0 > maxOff) ? maxOff : o0;
        unsigned l0 = ldsBase + (unsigned)tid * 16;
        asm volatile("global_load_async_to_lds_b128 %0, %1, %2"
                     :: "v"(l0), "v"(o0), "s"(gbase) : "memory");

        unsigned o1 = blockByte + (unsigned)(tid + 256) * 16;
        o1 = (o1 > maxOff) ? maxOff : o1;
        unsigned l1 = ldsBase + (unsigned)(tid + 256) * 16;
        asm volatile("global_load_async_to_lds_b128 %0, %1, %2"
                     :: "v"(l1), "v"(o1), "s"(gbase) : "memory");

        if (tid < 64) {
            unsigned o2 = blockByte + (unsigned)(tid + 512) * 16;
            o2 = (o2 > maxOff) ? maxOff : o2;
            unsigned l2 = ldsBase + (unsigned)(tid + 512) * 16;
            asm volatile("global_load_async_to_lds_b128 %0, %1, %2"
                         :: "v"(l2), "v"(o2), "s"(gbase) : "memory");
        }
    }

    // ---- overlap: init padded dist tiles to INF, mask tiles to 0 ----
    for (int i = tid; i < PROBS_PER_BLOCK * 196; i += THREADS) dpad[i] = PINF;
    for (int i = tid; i < PROBS_PER_BLOCK * WCELLS; i += THREADS) mlds[i] = 0;

    asm volatile("s_wait_asynccnt 0" ::: "memory");
    __syncthreads();

    // ---- per-lane row of costs via 5-entry table (select chain, stays in VGPRs) ----
    float cost[G], d[G];
    #pragma unroll
    for (int c = 0; c < G; ++c) { cost[c] = PINF; d[c] = PINF; }
    const float t0 = costs[0], t1 = costs[1], t2 = costs[2], t3 = costs[3], t4 = costs[4];
    if (active) {
        const int4* row = (const int4*)&stage[pLoc * WCELLS + r * G];
        int4 a = row[0], b = row[1], cc = row[2];
        int wv[G] = {a.x, a.y, a.z, a.w, b.x, b.y, b.z, b.w, cc.x, cc.y, cc.z, cc.w};
        #pragma unroll
        for (int c = 0; c < G; ++c) {
            int v = wv[c];
            cost[c] = (v == 0) ? t0 : (v == 1) ? t1 : (v == 2) ? t2 : (v == 3) ? t3 : t4;
        }
        if (r == 0) d[0] = cost[0];   // dist0[0,0] = cost[0,0]
    }

    // ---- min-plus Jacobi sweeps; rows in VGPRs, vertical exchange via ds_bpermute ----
    const int idxUp = ((lane - 1) & 31) << 2;   // wraps into the other problem's idle
    const int idxDn = ((lane + 1) & 31) << 2;   // INF lanes -> acts as border
    for (int sweep = 0; sweep < WCELLS; ++sweep) {
        float h2[G], h3[G];
        #pragma unroll
        for (int c = 0; c < G; ++c) {
            float lft = (c > 0)     ? d[c - 1] : PINF;
            float rgt = (c < G - 1) ? d[c + 1] : PINF;
            h2[c] = fminf(lft, rgt);            // horizontal, self excluded
            h3[c] = fminf(h2[c], d[c]);         // horizontal, self included (for up/down rows)
        }
        // issue all permutes back-to-back; consume in order (DS returns in order)
        float up[G], dn[G];
        #pragma unroll
        for (int c = 0; c < G; ++c)
            up[c] = __int_as_float(__builtin_amdgcn_ds_bpermute(idxUp, __float_as_int(h3[c])));
        #pragma unroll
        for (int c = 0; c < G; ++c)
            dn[c] = __int_as_float(__builtin_amdgcn_ds_bpermute(idxDn, __float_as_int(h3[c])));
        bool changed = false;
        #pragma unroll
        for (int c = 0; c < G; ++c) {
            float nbr = fminf(h2[c], fminf(up[c], dn[c]));
            float nd  = fminf(d[c], cost[c] + nbr);
            changed |= (nd < d[c]);
            d[c] = active ? nd : PINF;          // idle lanes stay INF (border)
        }
        if (!__any((int)changed)) break;        // converged fixed point == 144-sweep result
    }

    // ---- dump dist into 14x14 INF-padded LDS tile ----
    if (active) {
        #pragma unroll
        for (int c = 0; c < G; ++c)
            dpad[pLoc * 196 + (r + 1) * 14 + (c + 1)] = d[c];
    }
    __syncthreads();

    // ---- backtrack: one thread per problem, first-min tie-break (DIRS order) ----
    if (tid < PROBS_PER_BLOCK && (blockIdx.x * PROBS_PER_BLOCK + tid) < K) {
        const float* dp = &dpad[tid * 196];
        int* mk = &mlds[tid * WCELLS];
        int pi = G - 1, pj = G - 1;
        mk[pi * G + pj] = 1;
        const int DI[8] = {-1, -1, -1, 0, 0, 1, 1, 1};
        const int DJ[8] = {-1,  0,  1,-1, 1,-1, 0, 1};
        for (int step = 0; step < WCELLS; ++step) {
            if (pi == 0 && pj == 0) break;
            float best = 3.0e9f; int bi = pi, bj = pj;
            #pragma unroll
            for (int n = 0; n < 8; ++n) {
                int ci = pi + DI[n], cj = pj + DJ[n];
                float v = dp[(ci + 1) * 14 + (cj + 1)];   // pad border == INF == invalid
                if (v < best) { best = v; bi = ci; bj = cj; }
            }
            pi = bi; pj = bj;
            mk[pi * G + pj] = 1;
        }
    }
    __syncthreads();

    // ---- coalesced 16B stores of the int32 mask ----
    if (active) {
        int4* orow = (int4*)&out[(long long)k * WCELLS + r * G];
        const int4* mrow = (const int4*)&mlds[pLoc * WCELLS + r * G];
        orow[0] = mrow[0]; orow[1] = mrow[1]; orow[2] = mrow[2];
    }
}

extern "C" void kernel_launch(void* const* d_in, const int* in_sizes, int n_in,
                              void* d_out, int out_size, void* d_ws, size_t ws_size,
                              hipStream_t stream) {
    (void)n_in; (void)out_size; (void)d_ws; (void)ws_size;
    const int*   w     = (const int*)d_in[0];
    const float* costs = (const float*)d_in[1];
    int*         out   = (int*)d_out;
    const int K = in_sizes[0] / WCELLS;                       // S*B = 8192
    const int blocks = (K + PROBS_PER_BLOCK - 1) / PROBS_PER_BLOCK;
    PathAbsorbing_kernel<<<dim3(blocks), dim3(THREADS), LDS_BYTES, stream>>>(w, costs, out, K);
}